// GATScorer_77584289235409
// MI455X (gfx1250) — compile-verified
//
#include <hip/hip_runtime.h>
#include <hip/hip_bf16.h>

#define NEG_SLOPE 0.2f
#define ORD_NEG_INF 0x007FFFFFu  /* fordu(-inf) */

typedef __attribute__((ext_vector_type(2))) float v2f;
typedef __attribute__((ext_vector_type(8))) float v8f;

// ---------- helpers: order-preserving float<->uint for atomicMax ----------
__device__ __forceinline__ unsigned fordu(float f) {
  unsigned u = __float_as_uint(f);
  return (u & 0x80000000u) ? ~u : (u | 0x80000000u);
}
__device__ __forceinline__ float uordf(unsigned u) {
  return (u & 0x80000000u) ? __uint_as_float(u & 0x7fffffffu)
                           : __uint_as_float(~u);
}

// ---------- generic fill ----------
__global__ void fill_u32(unsigned* __restrict__ p, unsigned v, int nwords) {
  int i = blockIdx.x * blockDim.x + threadIdx.x;
  if (i < nwords) p[i] = v;
}

// ---------- fp32 WMMA GEMM: Hout[M,Nc] = A[M,K] @ W[K,Nc] ----------
// One wave computes a 16x64 output tile (4 x v_wmma_f32_16x16x4_f32 accumulators,
// A-fragment reused across the 4 N-tiles). M%16==0, K%4==0, Nc%64==0 assumed.
__global__ __launch_bounds__(32)
void wmma_gemm_f32(const float* __restrict__ A, const float* __restrict__ W,
                   float* __restrict__ Hout, int M, int K, int Nc) {
  const int lane = threadIdx.x;            // 0..31, EXEC all-ones (no divergence)
  const int row0 = blockIdx.y * 16;
  const int col0 = blockIdx.x * 64;
  const int r    = lane & 15;
  const int khi  = (lane >> 4) * 2;        // lanes 16-31 hold K+2,K+3
  v8f acc0 = {}, acc1 = {}, acc2 = {}, acc3 = {};
  const float* Arow = A + (size_t)(row0 + r) * K;
  for (int k0 = 0; k0 < K; k0 += 4) {
    v2f a;
    a.x = Arow[k0 + khi + 0];
    a.y = Arow[k0 + khi + 1];
    const float* Wr0 = W + (size_t)(k0 + khi) * Nc + col0 + r;
    const float* Wr1 = Wr0 + Nc;
    v2f b0, b1, b2, b3;
    b0.x = Wr0[0];  b0.y = Wr1[0];
    b1.x = Wr0[16]; b1.y = Wr1[16];
    b2.x = Wr0[32]; b2.y = Wr1[32];
    b3.x = Wr0[48]; b3.y = Wr1[48];
    acc0 = __builtin_amdgcn_wmma_f32_16x16x4_f32(false, a, false, b0, (short)0, acc0, false, false);
    acc1 = __builtin_amdgcn_wmma_f32_16x16x4_f32(false, a, false, b1, (short)0, acc1, false, false);
    acc2 = __builtin_amdgcn_wmma_f32_16x16x4_f32(false, a, false, b2, (short)0, acc2, false, false);
    acc3 = __builtin_amdgcn_wmma_f32_16x16x4_f32(false, a, false, b3, (short)0, acc3, false, false);
  }
  // C/D layout: VGPR v -> row (v) for lanes 0-15, row (8+v) for lanes 16-31; col = lane&15
  const int mbase = row0 + ((lane >> 4) ? 8 : 0);
  const int nidx  = col0 + r;
#pragma unroll
  for (int v = 0; v < 8; ++v) {
    float* orow = Hout + (size_t)(mbase + v) * Nc + nidx;
    orow[0]  = acc0[v];
    orow[16] = acc1[v];
    orow[32] = acc2[v];
    orow[48] = acc3[v];
  }
}

// ---------- per-(node,head) attention coefficients s,d ----------
__global__ void attn_sd_kernel(const float* __restrict__ h,
                               const float* __restrict__ a_src,
                               const float* __restrict__ a_dst,
                               float* __restrict__ s, float* __restrict__ d,
                               int n, int H, int C) {
  int idx = blockIdx.x * blockDim.x + threadIdx.x;
  if (idx >= n * H) return;
  int node = idx / H, hd = idx - node * H;
  const float* hp = h + (size_t)node * H * C + (size_t)hd * C;
  const float* as = a_src + hd * C;
  const float* ad = a_dst + hd * C;
  float ss = 0.f, dd = 0.f;
  for (int c = 0; c < C; c += 4) {
    float4 hv = *(const float4*)(hp + c);
    float4 av = *(const float4*)(as + c);
    float4 bv = *(const float4*)(ad + c);
    ss += hv.x*av.x + hv.y*av.y + hv.z*av.z + hv.w*av.w;
    dd += hv.x*bv.x + hv.y*bv.y + hv.z*bv.z + hv.w*bv.w;
  }
  s[idx] = ss;
  d[idx] = dd;
}

__device__ __forceinline__ void edge_sd(const int* __restrict__ ei, int e, int E,
                                        int& src, int& dst) {
  if (e < E) { src = ei[e]; dst = ei[E + e]; }
  else       { src = dst = e - E; }  // implicit self loop
}

// ---------- edge pass 1: segment max of leaky_relu(s[src]+d[dst]) ----------
__global__ void edge_max_kernel(const int* __restrict__ ei, int E, int n, int H,
                                const float* __restrict__ s, const float* __restrict__ d,
                                unsigned* __restrict__ emax) {
  int idx = blockIdx.x * blockDim.x + threadIdx.x;
  int EA = E + n;
  if (idx >= EA * H) return;
  int e = idx / H, h = idx - e * H;
  int src, dst; edge_sd(ei, e, E, src, dst);
  float v = s[src * H + h] + d[dst * H + h];
  v = (v > 0.f) ? v : NEG_SLOPE * v;
  atomicMax(&emax[dst * H + h], fordu(v));
}

// ---------- edge pass 2: exp(e - max) and segment denom ----------
__global__ void edge_exp_kernel(const int* __restrict__ ei, int E, int n, int H,
                                const float* __restrict__ s, const float* __restrict__ d,
                                const unsigned* __restrict__ emax,
                                float* __restrict__ denom, float* __restrict__ exbuf) {
  int idx = blockIdx.x * blockDim.x + threadIdx.x;
  int EA = E + n;
  if (idx >= EA * H) return;
  int e = idx / H, h = idx - e * H;
  int src, dst; edge_sd(ei, e, E, src, dst);
  float v = s[src * H + h] + d[dst * H + h];
  v = (v > 0.f) ? v : NEG_SLOPE * v;
  float ex = __expf(v - uordf(emax[dst * H + h]));
  exbuf[idx] = ex;
  atomicAdd(&denom[dst * H + h], ex);
}

// ---------- edge pass 3: weighted scatter-add of h[src] ----------
__global__ void edge_aggr_kernel(const int* __restrict__ ei, int E, int n, int H, int C,
                                 const float* __restrict__ hsrc,
                                 const float* __restrict__ exbuf,
                                 const float* __restrict__ denom,
                                 float* __restrict__ outb) {
  int idx = blockIdx.x * blockDim.x + threadIdx.x;
  int cpv = (H * C) >> 2;                 // float4 chunks per edge
  int EA = E + n;
  if (idx >= EA * cpv) return;
  int e  = idx / cpv;
  int c4 = (idx - e * cpv) << 2;
  int src, dst; edge_sd(ei, e, E, src, dst);
  int h = c4 / C;
  float alpha = exbuf[e * H + h] / (denom[dst * H + h] + 1e-16f);
  float4 hv = *(const float4*)(hsrc + (size_t)src * H * C + c4);
  float* op = outb + (size_t)dst * H * C + c4;
  atomicAdd(op + 0, hv.x * alpha);
  atomicAdd(op + 1, hv.y * alpha);
  atomicAdd(op + 2, hv.z * alpha);
  atomicAdd(op + 3, hv.w * alpha);
}

// ---------- bias + ELU (in-place) ----------
__global__ void bias_elu_kernel(float* __restrict__ outb, const float* __restrict__ bias,
                                int n, int HC) {
  int i = blockIdx.x * blockDim.x + threadIdx.x;
  if (i >= n * HC) return;
  int c = i % HC;
  float v = outb[i] + bias[c];
  outb[i] = (v > 0.f) ? v : (__expf(v) - 1.f);
}

// ---------- Set2Set: per-graph LSTM step (d=128, 4*d=512 gates) ----------
__global__ __launch_bounds__(128)
void s2s_lstm(const float* __restrict__ Wih, const float* __restrict__ Whh,
              const float* __restrict__ bih, const float* __restrict__ bhh,
              float* __restrict__ qstar, float* __restrict__ hbuf, float* __restrict__ cbuf) {
  int g = blockIdx.x, t = threadIdx.x;   // t in 0..127
  __shared__ float qs[256];
  __shared__ float hs[128];
  qs[t]       = qstar[g * 256 + t];
  qs[t + 128] = qstar[g * 256 + 128 + t];
  hs[t]       = hbuf[g * 128 + t];
  __syncthreads();
  float gate[4];
#pragma unroll
  for (int gi = 0; gi < 4; ++gi) {       // torch gate order i,f,g,o
    int row = gi * 128 + t;
    const float* wi = Wih + (size_t)row * 256;
    const float* wh = Whh + (size_t)row * 128;
    float acc = bih[row] + bhh[row];
    for (int k = 0; k < 256; ++k) acc += wi[k] * qs[k];
    for (int k = 0; k < 128; ++k) acc += wh[k] * hs[k];
    gate[gi] = acc;
  }
  float ig = 1.f / (1.f + __expf(-gate[0]));
  float fg = 1.f / (1.f + __expf(-gate[1]));
  float gg = tanhf(gate[2]);
  float og = 1.f / (1.f + __expf(-gate[3]));
  float c = fg * cbuf[g * 128 + t] + ig * gg;
  float h = og * tanhf(c);
  cbuf[g * 128 + t] = c;
  hbuf[g * 128 + t] = h;
  qstar[g * 256 + t] = h;                // q part of q_star
}

// ---------- Set2Set: reset r half of q_star + per-graph max/denom ----------
__global__ void s2s_step_init(float* __restrict__ qstar, unsigned* __restrict__ emaxg,
                              float* __restrict__ denomg, int B) {
  int i = blockIdx.x * blockDim.x + threadIdx.x;
  if (i < B * 128) {
    int g = i >> 7, c = i & 127;
    qstar[g * 256 + 128 + c] = 0.f;
  }
  if (i < B) { emaxg[i] = ORD_NEG_INF; denomg[i] = 0.f; }
}

// ---------- Set2Set: e[n] = x[n].q[batch[n]], segment max ----------
__global__ void s2s_dot_max(const float* __restrict__ x, const int* __restrict__ batch,
                            const float* __restrict__ qstar, float* __restrict__ enode,
                            unsigned* __restrict__ emaxg, int n) {
  int node = blockIdx.x * blockDim.x + threadIdx.x;
  if (node >= n) return;
  int g = batch[node];
  const float* xp = x + (size_t)node * 128;
  const float* qp = qstar + (size_t)g * 256;   // q in first half
  float e = 0.f;
  for (int c = 0; c < 128; c += 4) {
    float4 xv = *(const float4*)(xp + c);
    float4 qv = *(const float4*)(qp + c);
    e += xv.x*qv.x + xv.y*qv.y + xv.z*qv.z + xv.w*qv.w;
  }
  enode[node] = e;
  atomicMax(&emaxg[g], fordu(e));
}

// ---------- Set2Set: numerator + segment denom ----------
__global__ void s2s_exp(const int* __restrict__ batch, const float* __restrict__ enode,
                        const unsigned* __restrict__ emaxg, float* __restrict__ denomg,
                        float* __restrict__ anode, int n) {
  int node = blockIdx.x * blockDim.x + threadIdx.x;
  if (node >= n) return;
  int g = batch[node];
  float a = __expf(enode[node] - uordf(emaxg[g]));
  anode[node] = a;
  atomicAdd(&denomg[g], a);
}

// ---------- Set2Set: r = segsum(a*x) accumulated into q_star second half ----------
__global__ void s2s_aggr(const float* __restrict__ x, const int* __restrict__ batch,
                         const float* __restrict__ anode, const float* __restrict__ denomg,
                         float* __restrict__ qstar, int n) {
  int idx = blockIdx.x * blockDim.x + threadIdx.x;   // n * 32 chunks of float4
  if (idx >= n * 32) return;
  int node = idx >> 5;
  int c4   = (idx & 31) << 2;
  int g = batch[node];
  float coef = anode[node] / (denomg[g] + 1e-16f);
  float4 xv = *(const float4*)(x + (size_t)node * 128 + c4);
  float* rp = qstar + (size_t)g * 256 + 128 + c4;
  atomicAdd(rp + 0, xv.x * coef);
  atomicAdd(rp + 1, xv.y * coef);
  atomicAdd(rp + 2, xv.z * coef);
  atomicAdd(rp + 3, xv.w * coef);
}

// ---------- final head: gf-MLP + concat + scorer MLP, one block per graph ----------
__global__ __launch_bounds__(128)
void final_head(const float* __restrict__ qstar, const float* __restrict__ gfeat,
                const float* __restrict__ gW1, const float* __restrict__ gb1,
                const float* __restrict__ gW2, const float* __restrict__ gb2,
                const float* __restrict__ mW1, const float* __restrict__ mb1,
                const float* __restrict__ mW2, const float* __restrict__ mb2,
                float* __restrict__ out) {
  int g = blockIdx.x, t = threadIdx.x;
  __shared__ float z[288];
  __shared__ float g1[64];
  __shared__ float m1[128];
  z[t]       = qstar[g * 256 + t];
  z[t + 128] = qstar[g * 256 + 128 + t];
  if (t < 64) {                          // hidden1 = relu(gf @ gW1 + gb1)
    float acc = gb1[t];
    for (int k = 0; k < 9; ++k) acc += gfeat[g * 9 + k] * gW1[k * 64 + t];
    g1[t] = acc > 0.f ? acc : 0.f;
  }
  __syncthreads();
  if (t < 32) {                          // gf2 = hidden1 @ gW2 + gb2
    float acc = gb2[t];
    for (int k = 0; k < 64; ++k) acc += g1[k] * gW2[k * 32 + t];
    z[256 + t] = acc;
  }
  __syncthreads();
  {                                      // m1 = relu(z @ mW1 + mb1), 288 -> 128
    float acc = mb1[t];
    for (int k = 0; k < 288; ++k) acc += z[k] * mW1[k * 128 + t];
    m1[t] = acc > 0.f ? acc : 0.f;
  }
  __syncthreads();
  if (t < 4) {                           // scores = m1 @ mW2 + mb2
    float acc = mb2[t];
    for (int k = 0; k < 128; ++k) acc += m1[k] * mW2[k * 4 + t];
    out[g * 4 + t] = acc;
  }
}

// =====================================================================

extern "C" void kernel_launch(void* const* d_in, const int* in_sizes, int n_in,
                              void* d_out, int out_size, void* d_ws, size_t ws_size,
                              hipStream_t stream) {
  const float* x      = (const float*)d_in[0];
  const int*   ei     = (const int*)d_in[1];
  const int*   batch  = (const int*)d_in[2];
  const float* gfeat  = (const float*)d_in[3];
  const float* W1     = (const float*)d_in[4];
  const float* asrc1  = (const float*)d_in[5];
  const float* adst1  = (const float*)d_in[6];
  const float* b1     = (const float*)d_in[7];
  const float* W2     = (const float*)d_in[8];
  const float* asrc2  = (const float*)d_in[9];
  const float* adst2  = (const float*)d_in[10];
  const float* b2     = (const float*)d_in[11];
  const float* W3     = (const float*)d_in[12];
  const float* asrc3  = (const float*)d_in[13];
  const float* adst3  = (const float*)d_in[14];
  const float* b3     = (const float*)d_in[15];
  const float* Wih    = (const float*)d_in[16];
  const float* Whh    = (const float*)d_in[17];
  const float* bih    = (const float*)d_in[18];
  const float* bhh    = (const float*)d_in[19];
  const float* gW1    = (const float*)d_in[20];
  const float* gb1    = (const float*)d_in[21];
  const float* gW2    = (const float*)d_in[22];
  const float* gb2    = (const float*)d_in[23];
  const float* mW1    = (const float*)d_in[24];
  const float* mb1    = (const float*)d_in[25];
  const float* mW2    = (const float*)d_in[26];
  const float* mb2    = (const float*)d_in[27];
  float* scores = (float*)d_out;

  const int N  = in_sizes[0] / 128;   // 20000
  const int E  = in_sizes[1] / 2;     // 320000
  const int B  = in_sizes[3] / 9;     // 64
  const int EA = E + N;

  // workspace layout (floats)
  float* ws   = (float*)d_ws;
  float* hA   = ws;                         // [N,256] GEMM output
  float* hB   = hA + (size_t)N * 256;       // [N,256] aggregated / layer output
  float* sbuf = hB + (size_t)N * 256;       // [N,4]
  float* dbuf = sbuf + (size_t)N * 4;       // [N,4]
  unsigned* emaxI = (unsigned*)(dbuf + (size_t)N * 4);  // [N,4]
  float* denom = (float*)(emaxI + (size_t)N * 4);       // [N,4]
  float* exbuf = denom + (size_t)N * 4;     // [EA,4]
  float* lstm_h = exbuf + (size_t)EA * 4;   // [B,128]
  float* lstm_c = lstm_h + (size_t)B * 128; // [B,128]
  float* qstar  = lstm_c + (size_t)B * 128; // [B,256]
  float* enode  = qstar + (size_t)B * 256;  // [N]
  float* anode  = enode + N;                // [N]
  unsigned* emaxg = (unsigned*)(anode + N); // [B]
  float* denomg   = (float*)(emaxg + B);    // [B]

  const int TB = 256;
  auto blocks = [](long n, int tb) { return (unsigned)((n + tb - 1) / tb); };

  // ---------------- one GAT layer ----------------
  auto run_gat = [&](const float* in, int K, const float* W, const float* as,
                     const float* ad, const float* bias, int H, int C, float* outb) {
    const int HC = H * C;
    dim3 gg(HC / 64, N / 16);
    wmma_gemm_f32<<<gg, 32, 0, stream>>>(in, W, hA, N, K, HC);
    attn_sd_kernel<<<blocks((long)N * H, TB), TB, 0, stream>>>(hA, as, ad, sbuf, dbuf, N, H, C);
    fill_u32<<<blocks((long)N * H, TB), TB, 0, stream>>>(emaxI, ORD_NEG_INF, N * H);
    fill_u32<<<blocks((long)N * H, TB), TB, 0, stream>>>((unsigned*)denom, 0u, N * H);
    fill_u32<<<blocks((long)N * HC, TB), TB, 0, stream>>>((unsigned*)outb, 0u, N * HC);
    edge_max_kernel<<<blocks((long)EA * H, TB), TB, 0, stream>>>(ei, E, N, H, sbuf, dbuf, emaxI);
    edge_exp_kernel<<<blocks((long)EA * H, TB), TB, 0, stream>>>(ei, E, N, H, sbuf, dbuf, emaxI, denom, exbuf);
    edge_aggr_kernel<<<blocks((long)EA * (HC / 4), TB), TB, 0, stream>>>(ei, E, N, H, C, hA, exbuf, denom, outb);
    bias_elu_kernel<<<blocks((long)N * HC, TB), TB, 0, stream>>>(outb, bias, N, HC);
  };

  run_gat(x,  128, W1, asrc1, adst1, b1, 4, 64,  hB);   // layer 1: 128 -> 4x64
  run_gat(hB, 256, W2, asrc2, adst2, b2, 4, 64,  hB);   // layer 2: 256 -> 4x64
  run_gat(hB, 256, W3, asrc3, adst3, b3, 1, 128, hB);   // layer 3: 256 -> 128

  // ---------------- Set2Set (d = 128, 3 steps) ----------------
  {
    int zwords = B * 128 * 2 + B * 256;   // lstm_h, lstm_c, qstar contiguous
    fill_u32<<<blocks(zwords, TB), TB, 0, stream>>>((unsigned*)lstm_h, 0u, zwords);
  }
  for (int step = 0; step < 3; ++step) {
    s2s_lstm<<<B, 128, 0, stream>>>(Wih, Whh, bih, bhh, qstar, lstm_h, lstm_c);
    s2s_step_init<<<blocks((long)B * 128, TB), TB, 0, stream>>>(qstar, emaxg, denomg, B);
    s2s_dot_max<<<blocks(N, TB), TB, 0, stream>>>(hB, batch, qstar, enode, emaxg, N);
    s2s_exp<<<blocks(N, TB), TB, 0, stream>>>(batch, enode, emaxg, denomg, anode, N);
    s2s_aggr<<<blocks((long)N * 32, TB), TB, 0, stream>>>(hB, batch, anode, denomg, qstar, N);
  }

  // ---------------- final MLP head ----------------
  final_head<<<B, 128, 0, stream>>>(qstar, gfeat, gW1, gb1, gW2, gb2,
                                    mW1, mb1, mW2, mb2, scores);
}